// ConvCaps_9337258901601
// MI455X (gfx1250) — compile-verified
//
#include <hip/hip_runtime.h>
#include <hip/hip_bf16.h>
#include <math.h>

typedef float v2f __attribute__((ext_vector_type(2)));
typedef float v8f __attribute__((ext_vector_type(8)));

#define NCAPS_A 32
#define NCAPS_B 32
#define KKC     9
#define KKA     288
#define NX      512     // b * oh * ow = 8*64
#define NCOL    512     // B * PSIZE
#define BN_EPS  1e-5f

// ---------------------------------------------------------------------------
// Kernel 0a: fold W_caps with Wk / Wv into per-(n,o) 16x16 matrices.
// TK[no][m][p=i*4+j] = sum_k Wk[m][i*4+k] * Wc[no][j][k]   (same for TV/Wv)
// ---------------------------------------------------------------------------
__global__ void build_T_kernel(const float* __restrict__ Wc,
                               const float* __restrict__ Wk,
                               const float* __restrict__ Wv,
                               float* __restrict__ TK,
                               float* __restrict__ TV,
                               int total)
{
    int t = blockIdx.x * blockDim.x + threadIdx.x;
    if (t >= total) return;
    int pij = t & 15;
    int m   = (t >> 4) & 15;
    int no  = t >> 8;                 // n*32 + o, 0..9215
    int i = pij >> 2, j = pij & 3;
    const float* wc = Wc + (size_t)no * 16 + j * 4;
    const float* wk = Wk + m * 16 + i * 4;
    const float* wv = Wv + m * 16 + i * 4;
    float sk = 0.f, sv = 0.f;
#pragma unroll
    for (int k = 0; k < 4; ++k) {
        sk += wk[k] * wc[k];
        sv += wv[k] * wc[k];
    }
    TK[t] = sk;
    TV[t] = sv;
}

// ---------------------------------------------------------------------------
// Kernel 0b: im2col / unfold into poseT[n][p][x]  (B-operand friendly layout)
// poseT[(n*16+p)*512 + x] = x_in[b, a*16+p, 2*r+kh, 2*c+kw]
//   n = kk*32 + a, kk = kh*3+kw, x = b*64 + r*8 + c
// ---------------------------------------------------------------------------
__global__ void build_pose_kernel(const float* __restrict__ X,
                                  float* __restrict__ poseT,
                                  int total)
{
    int t = blockIdx.x * blockDim.x + threadIdx.x;
    if (t >= total) return;
    int x = t & 511;
    int p = (t >> 9) & 15;
    int n = t >> 13;                  // 0..287
    int a  = n & 31;
    int kk = n >> 5;
    int kh = kk / 3, kw = kk % 3;
    int b  = x >> 6;
    int s  = x & 63;
    int r  = s >> 3, c = s & 7;
    size_t src = ((size_t)(b * 512 + a * 16 + p) * 17 + (2 * r + kh)) * 17 + (2 * c + kw);
    poseT[t] = X[src];
}

// ---------------------------------------------------------------------------
// Kernel 1: fused  Kt/Vt GEMM (WMMA f32)  +  gaussian ll  +  online softmax
//           over n=288  +  weighted Vt accumulation.
// Block = 1 wave (32 lanes); grid = (xtile 0..31) x (o 0..31).
// Lane l (lo=l&15, hi=l>>4) owns x = xbase+lo, rows m = hi*8 .. hi*8+7
// (heads 2*hi and 2*hi+1) — fully register local.
// ---------------------------------------------------------------------------
__global__ __launch_bounds__(32)
void caps_attn_kernel(const float* __restrict__ TK,
                      const float* __restrict__ TV,
                      const float* __restrict__ poseT,
                      const float* __restrict__ Mmu,
                      const float* __restrict__ Ssig,
                      const float* __restrict__ p_logit,
                      const float* __restrict__ bk,
                      const float* __restrict__ bv,
                      float* __restrict__ O0)
{
    const int o    = blockIdx.x & 31;
    const int xt   = blockIdx.x >> 5;
    const int lane = threadIdx.x;
    const int lo   = lane & 15;
    const int hi   = lane >> 4;
    const int xbase = xt * 16;

    // per-lane row constants for rows m = hi*8 + r
    float mu8[8], isg8[8], ct8[8], bk8[8], bv8[8];
#pragma unroll
    for (int r = 0; r < 8; ++r) {
        int m = hi * 8 + r;
        float sv = Ssig[o * 16 + m];
        float sg = (sv > 20.f) ? sv : log1pf(__expf(sv));   // softplus
        mu8[r]  = Mmu[o * 16 + m];
        isg8[r] = 1.0f / sg;
        ct8[r]  = -0.9189385332046727f - logf(sg);          // -0.5*ln(2pi) - ln(sigma)
        bk8[r]  = bk[m];
        bv8[r]  = bv[m];
    }
    // logp[o] = p_logit[o] - logsumexp(p_logit)
    float pmax = -INFINITY;
    for (int i = 0; i < 32; ++i) pmax = fmaxf(pmax, p_logit[i]);
    float psum = 0.f;
    for (int i = 0; i < 32; ++i) psum += __expf(p_logit[i] - pmax);
    float logp = p_logit[o] - (pmax + logf(psum));

    // online softmax state (two heads per lane)
    float mrun0 = -INFINITY, mrun1 = -INFINITY;
    float srun0 = 0.f, srun1 = 0.f;
    float acc[8];
#pragma unroll
    for (int r = 0; r < 8; ++r) acc[r] = 0.f;

    for (int n = 0; n < KKA; ++n) {
        const float* tk = TK + (size_t)(n * 32 + o) * 256;
        const float* tv = TV + (size_t)(n * 32 + o) * 256;
        const float* pb = poseT + (size_t)n * (16 * NX) + xbase + lo;

        v2f ak[4], av[4], bf[4];
#pragma unroll
        for (int kc = 0; kc < 4; ++kc) {
            int cb = 4 * kc + 2 * hi;
            ak[kc][0] = tk[lo * 16 + cb];
            ak[kc][1] = tk[lo * 16 + cb + 1];
            av[kc][0] = tv[lo * 16 + cb];
            av[kc][1] = tv[lo * 16 + cb + 1];
            bf[kc][0] = pb[(size_t)(cb + 0) * NX];
            bf[kc][1] = pb[(size_t)(cb + 1) * NX];
        }
        v8f ck = {}; v8f cv = {};
#pragma unroll
        for (int kc = 0; kc < 4; ++kc) {
            ck = __builtin_amdgcn_wmma_f32_16x16x4_f32(false, ak[kc], false, bf[kc],
                                                       (short)0, ck, false, false);
            cv = __builtin_amdgcn_wmma_f32_16x16x4_f32(false, av[kc], false, bf[kc],
                                                       (short)0, cv, false, false);
        }
        // gaussian ll (sum over d per head) + Vt with bias
        float ll0 = 0.f, ll1 = 0.f;
        float vt[8];
#pragma unroll
        for (int r = 0; r < 8; ++r) {
            float kt = ck[r] + bk8[r];
            float d  = (kt - mu8[r]) * isg8[r];
            float term = ct8[r] - 0.5f * d * d;
            if (r < 4) ll0 += term; else ll1 += term;
            vt[r] = cv[r] + bv8[r];
        }
        float z0 = (ll0 + logp) * 0.25f;   // / sqrt(dim=16)
        float z1 = (ll1 + logp) * 0.25f;

        float mn0 = fmaxf(mrun0, z0);
        float al0 = __expf(mrun0 - mn0);
        float w0  = __expf(z0 - mn0);
        srun0 = srun0 * al0 + w0;  mrun0 = mn0;

        float mn1 = fmaxf(mrun1, z1);
        float al1 = __expf(mrun1 - mn1);
        float w1  = __expf(z1 - mn1);
        srun1 = srun1 * al1 + w1;  mrun1 = mn1;

#pragma unroll
        for (int r = 0; r < 4; ++r) acc[r] = acc[r] * al0 + w0 * vt[r];
#pragma unroll
        for (int r = 4; r < 8; ++r) acc[r] = acc[r] * al1 + w1 * vt[r];
    }

    int x = xbase + lo;
#pragma unroll
    for (int r = 0; r < 8; ++r) {
        int m = hi * 8 + r;
        float s = (r < 4) ? srun0 : srun1;
        O0[(size_t)x * NCOL + o * 16 + m] = mu8[r] + acc[r] / s;
    }
}

// ---------------------------------------------------------------------------
// Kernel 2/4: per-column mean & variance over the 512-row axis (batch norm)
// ---------------------------------------------------------------------------
__global__ __launch_bounds__(128)
void colstats_kernel(const float* __restrict__ Buf,
                     float* __restrict__ mean,
                     float* __restrict__ var)
{
    __shared__ float ssum[128], ssq[128];
    int col = blockIdx.x;
    int tid = threadIdx.x;
    float s = 0.f, q = 0.f;
    for (int row = tid; row < NX; row += 128) {
        float v = Buf[(size_t)row * NCOL + col];
        s += v; q += v * v;
    }
    ssum[tid] = s; ssq[tid] = q;
    __syncthreads();
    for (int off = 64; off > 0; off >>= 1) {
        if (tid < off) { ssum[tid] += ssum[tid + off]; ssq[tid] += ssq[tid + off]; }
        __syncthreads();
    }
    if (tid == 0) {
        float mu = ssum[0] * (1.0f / NX);
        mean[col] = mu;
        var[col]  = ssq[0] * (1.0f / NX) - mu * mu;
    }
}

// ---------------------------------------------------------------------------
// Kernel 3: bn0 apply + residual MLP  Y = On + relu(On @ Wo.T + bo)  (WMMA)
// ---------------------------------------------------------------------------
__global__ __launch_bounds__(32)
void mlp_kernel(const float* __restrict__ O0,
                const float* __restrict__ Wo,
                const float* __restrict__ bo,
                const float* __restrict__ g0,
                const float* __restrict__ beta0,
                const float* __restrict__ mean0,
                const float* __restrict__ var0,
                float* __restrict__ Y)
{
    __shared__ float OnT[16][17];     // [k][x], padded
    const int o    = blockIdx.x & 31;
    const int xt   = blockIdx.x >> 5;
    const int lane = threadIdx.x;
    const int lo   = lane & 15;
    const int hi   = lane >> 4;
    const int xbase = xt * 16;

#pragma unroll
    for (int r = 0; r < 8; ++r) {
        int k = hi * 8 + r;
        int col = o * 16 + k;
        float v  = O0[(size_t)(xbase + lo) * NCOL + col];
        float on = g0[col] * (v - mean0[col]) * rsqrtf(var0[col] + BN_EPS) + beta0[col];
        OnT[k][lo] = on;
    }
    __syncthreads();

    v2f af[4], bf[4];
#pragma unroll
    for (int kc = 0; kc < 4; ++kc) {
        int cb = 4 * kc + 2 * hi;
        af[kc][0] = Wo[lo * 16 + cb];
        af[kc][1] = Wo[lo * 16 + cb + 1];
        bf[kc][0] = OnT[cb + 0][lo];
        bf[kc][1] = OnT[cb + 1][lo];
    }
    v8f cf = {};
#pragma unroll
    for (int kc = 0; kc < 4; ++kc)
        cf = __builtin_amdgcn_wmma_f32_16x16x4_f32(false, af[kc], false, bf[kc],
                                                   (short)0, cf, false, false);
#pragma unroll
    for (int r = 0; r < 8; ++r) {
        int m = hi * 8 + r;
        float rel = fmaxf(cf[r] + bo[m], 0.f);
        Y[(size_t)(xbase + lo) * NCOL + o * 16 + m] = OnT[m][lo] + rel;
    }
}

// ---------------------------------------------------------------------------
// Kernel 5: bn1 apply + layout permute (512,512) -> (8, 512, 8, 8)
// ---------------------------------------------------------------------------
__global__ void final_kernel(const float* __restrict__ Y,
                             const float* __restrict__ g1,
                             const float* __restrict__ beta1,
                             const float* __restrict__ mean1,
                             const float* __restrict__ var1,
                             float* __restrict__ out,
                             int total)
{
    int t = blockIdx.x * blockDim.x + threadIdx.x;
    if (t >= total) return;
    int c  = t & 7;
    int r  = (t >> 3) & 7;
    int cf = (t >> 6) & 511;
    int b  = t >> 15;
    int x  = b * 64 + r * 8 + c;
    float v = Y[(size_t)x * NCOL + cf];
    out[t] = g1[cf] * (v - mean1[cf]) * rsqrtf(var1[cf] + BN_EPS) + beta1[cf];
}

// ---------------------------------------------------------------------------
extern "C" void kernel_launch(void* const* d_in, const int* in_sizes, int n_in,
                              void* d_out, int out_size, void* d_ws, size_t ws_size,
                              hipStream_t stream)
{
    const float* x_in   = (const float*)d_in[0];
    const float* W_caps = (const float*)d_in[1];
    const float* Mmu    = (const float*)d_in[2];
    const float* Ssig   = (const float*)d_in[3];
    const float* p_log  = (const float*)d_in[4];
    const float* Wk     = (const float*)d_in[5];
    const float* bk     = (const float*)d_in[6];
    const float* Wv     = (const float*)d_in[7];
    const float* bv     = (const float*)d_in[8];
    const float* Wo     = (const float*)d_in[9];
    const float* bo     = (const float*)d_in[10];
    const float* g0     = (const float*)d_in[11];
    const float* beta0  = (const float*)d_in[12];
    const float* g1     = (const float*)d_in[13];
    const float* beta1  = (const float*)d_in[14];
    float* out = (float*)d_out;

    float* w = (float*)d_ws;
    float* TK    = w;  w += 9216 * 256;      // 2,359,296
    float* TV    = w;  w += 9216 * 256;      // 2,359,296
    float* poseT = w;  w += KKA * 16 * NX;   // 2,359,296
    float* O0    = w;  w += NX * NCOL;       //   262,144
    float* Y     = w;  w += NX * NCOL;       //   262,144
    float* mean0 = w;  w += 512;
    float* var0  = w;  w += 512;
    float* mean1 = w;  w += 512;
    float* var1  = w;  w += 512;

    const int tT = 9216 * 256;
    build_T_kernel<<<(tT + 255) / 256, 256, 0, stream>>>(W_caps, Wk, Wv, TK, TV, tT);

    const int tP = KKA * 16 * NX;
    build_pose_kernel<<<(tP + 255) / 256, 256, 0, stream>>>(x_in, poseT, tP);

    caps_attn_kernel<<<1024, 32, 0, stream>>>(TK, TV, poseT, Mmu, Ssig, p_log, bk, bv, O0);

    colstats_kernel<<<512, 128, 0, stream>>>(O0, mean0, var0);

    mlp_kernel<<<1024, 32, 0, stream>>>(O0, Wo, bo, g0, beta0, mean0, var0, Y);

    colstats_kernel<<<512, 128, 0, stream>>>(Y, mean1, var1);

    const int tO = 8 * 512 * 64;
    final_kernel<<<(tO + 255) / 256, 256, 0, stream>>>(Y, g1, beta1, mean1, var1, out, tO);
}